// GATDecoder_32959579030040
// MI455X (gfx1250) — compile-verified
//
#include <hip/hip_runtime.h>

typedef __attribute__((ext_vector_type(2))) float v2f;
typedef __attribute__((ext_vector_type(8))) float v8f;

#define NEG_SLOPE 0.2f

// ---------- monotone float <-> u32 encoding for atomic max ----------
__device__ __forceinline__ unsigned enc_f32(float f) {
  unsigned u = __float_as_uint(f);
  return (u & 0x80000000u) ? ~u : (u | 0x80000000u);
}
__device__ __forceinline__ float dec_f32(unsigned u) {
  unsigned b = (u & 0x80000000u) ? (u & 0x7fffffffu) : ~u;
  return __uint_as_float(b);
}
// enc_f32(-inf) == 0x007FFFFF
#define ENC_NEG_INF 0x007FFFFFu

// ---------- fills (ws/out are poisoned by harness; re-init each call) ----------
__global__ void fill_f32(float* __restrict__ p, float v, long long n) {
  long long i = (long long)blockIdx.x * blockDim.x + threadIdx.x;
  if (i < n) p[i] = v;
}
__global__ void fill_u32(unsigned* __restrict__ p, unsigned v, int n) {
  int i = blockIdx.x * blockDim.x + threadIdx.x;
  if (i < n) p[i] = v;
}

// ---------- FP32 WMMA GEMM: H[M,N] = X[M,K] @ W[K,N] ----------
// One wave32 per 16x64 output strip (4 N-tiles) so the A fragment is loaded
// once per K-step and feeds FOUR INDEPENDENT v_wmma accumulator chains
// (breaks the D->C RAW serialization and hides WMMA latency).
// Fragment layouts per CDNA5 ISA 7.12.2:
//   A 16x4 : lanes 0-15 rows, VGPR0/1 = K {0,1}; lanes 16-31 rows, K {2,3}
//   B 4x16 : symmetric (lane-half selects K pair, vector elem selects K within pair)
//   C/D    : VGPR v -> row v + 8*(lane>=16), col = lane&15
__global__ void gemm_f32_wmma(const float* __restrict__ X, const float* __restrict__ W,
                              float* __restrict__ H, int M, int K, int N) {
  int lane = threadIdx.x & 31;
  int wave = (int)((blockIdx.x * (long long)blockDim.x + threadIdx.x) >> 5);
  int nt4 = N >> 6;                     // strips of 4 tiles (N is 128 or 256)
  int mt  = (M + 15) >> 4;
  if (wave >= mt * nt4) return;         // wave-uniform: EXEC stays all-ones for WMMA
  int tm = wave / nt4, tn = wave % nt4;
  int half = lane >> 4;                 // 0: lanes 0-15, 1: lanes 16-31
  int l15  = lane & 15;
  int row = tm * 16 + l15;  if (row >= M) row = M - 1;   // clamp loads (tiles exact here)
  int col0 = tn * 64 + l15;
  v8f acc0 = {}, acc1 = {}, acc2 = {}, acc3 = {};
  for (int k0 = 0; k0 < K; k0 += 4) {
    int kb = k0 + (half << 1);
    v2f a = *(const v2f*)(X + (size_t)row * K + kb);     // contiguous K pair, shared x4
    const float* w0 = W + (size_t)kb * N;
    const float* w1 = W + (size_t)(kb + 1) * N;
    v2f b0, b1, b2, b3;
    b0.x = w0[col0 +  0]; b0.y = w1[col0 +  0];
    b1.x = w0[col0 + 16]; b1.y = w1[col0 + 16];
    b2.x = w0[col0 + 32]; b2.y = w1[col0 + 32];
    b3.x = w0[col0 + 48]; b3.y = w1[col0 + 48];
    acc0 = __builtin_amdgcn_wmma_f32_16x16x4_f32(false, a, false, b0, (short)0, acc0, false, false);
    acc1 = __builtin_amdgcn_wmma_f32_16x16x4_f32(false, a, false, b1, (short)0, acc1, false, false);
    acc2 = __builtin_amdgcn_wmma_f32_16x16x4_f32(false, a, false, b2, (short)0, acc2, false, false);
    acc3 = __builtin_amdgcn_wmma_f32_16x16x4_f32(false, a, false, b3, (short)0, acc3, false, false);
  }
  int rbase = tm * 16 + half * 8;
#pragma unroll
  for (int v = 0; v < 8; ++v) {
    int r = rbase + v;
    if (r < M) {
      float* hp = H + (size_t)r * N + col0;
      hp[ 0] = acc0[v];
      hp[16] = acc1[v];
      hp[32] = acc2[v];
      hp[48] = acc3[v];
    }
  }
}

// ---------- per-node attention logits: as[n] = h[n].a_src, ad[n] = h[n].a_dst ----------
__global__ void node_alpha(const float* __restrict__ h, const float* __restrict__ avs,
                           const float* __restrict__ avd, float* __restrict__ as,
                           float* __restrict__ ad, int Nn, int F) {
  int lane = threadIdx.x & 31;
  int node = (int)((blockIdx.x * (long long)blockDim.x + threadIdx.x) >> 5);
  if (node >= Nn) return;
  const float* hp = h + (size_t)node * F;
  float ss = 0.f, sd = 0.f;
  for (int c = lane; c < F; c += 32) {
    float hv = hp[c];
    ss += hv * avs[c];
    sd += hv * avd[c];
  }
#pragma unroll
  for (int off = 16; off > 0; off >>= 1) {
    ss += __shfl_xor(ss, off, 32);
    sd += __shfl_xor(sd, off, 32);
  }
  if (lane == 0) { as[node] = ss; ad[node] = sd; }
}

__device__ __forceinline__ void edge_endpoints(int t, int E, const int* __restrict__ esrc,
                                               const int* __restrict__ edst, int& s, int& d) {
  if (t < E) { s = esrc[t]; d = edst[t]; } else { s = t - E; d = s; }  // self-loops appended
}

// ---------- segment max of LeakyReLU(e) over dst ----------
__global__ void edge_max(const float* __restrict__ as, const float* __restrict__ ad,
                         const int* __restrict__ esrc, const int* __restrict__ edst,
                         unsigned* __restrict__ menc, int E, int Nn) {
  int t = blockIdx.x * blockDim.x + threadIdx.x;
  if (t >= E + Nn) return;
  int s, d; edge_endpoints(t, E, esrc, edst, s, d);
  float e = as[s] + ad[d];
  e = (e > 0.f) ? e : NEG_SLOPE * e;
  atomicMax(menc + d, enc_f32(e));
}

// ---------- w = exp(e - m[dst]); z[dst] += w ----------
__global__ void edge_expsum(const float* __restrict__ as, const float* __restrict__ ad,
                            const int* __restrict__ esrc, const int* __restrict__ edst,
                            const unsigned* __restrict__ menc, float* __restrict__ w,
                            float* __restrict__ z, int E, int Nn) {
  int t = blockIdx.x * blockDim.x + threadIdx.x;
  if (t >= E + Nn) return;
  int s, d; edge_endpoints(t, E, esrc, edst, s, d);
  float e = as[s] + ad[d];
  e = (e > 0.f) ? e : NEG_SLOPE * e;
  float wv = expf(e - dec_f32(menc[d]));
  w[t] = wv;
  atomicAdd(z + d, wv);
}

// ---------- acc[dst] += (w/z[dst]) * h[src] ; thread per (edge, 4 channels) ----------
__global__ void aggregate(const float* __restrict__ h, const int* __restrict__ esrc,
                          const int* __restrict__ edst, const float* __restrict__ w,
                          const float* __restrict__ z, float* __restrict__ acc,
                          int E, int Nn, int F) {
  long long t = (long long)blockIdx.x * blockDim.x + threadIdx.x;
  int f4 = F >> 2;
  long long total = (long long)(E + Nn) * f4;
  if (t >= total) return;
  int eix = (int)(t / f4);
  int c = ((int)(t % f4)) << 2;
  int s, d; edge_endpoints(eix, E, esrc, edst, s, d);
  float alpha = w[eix] / z[d];
  const float4 hv = *(const float4*)(h + (size_t)s * F + c);
  float* op = acc + (size_t)d * F + c;
  atomicAdd(op + 0, alpha * hv.x);
  atomicAdd(op + 1, alpha * hv.y);
  atomicAdd(op + 2, alpha * hv.z);
  atomicAdd(op + 3, alpha * hv.w);
}

// ---------- out = acc + b (optional ReLU) ----------
__global__ void bias_act(const float* __restrict__ acc, const float* __restrict__ b,
                         float* __restrict__ out, long long total, int F, int do_relu) {
  long long t = (long long)blockIdx.x * blockDim.x + threadIdx.x;
  if (t >= total) return;
  float v = acc[t] + b[(int)(t % F)];
  out[t] = do_relu ? fmaxf(v, 0.f) : v;
}

// ---------------------------------------------------------------------------
static void run_layer(const float* xin, int Fin, const float* W,
                      const float* avs, const float* avd, const float* bias,
                      float* h, float* accb, float* xout, int Fout, int do_relu,
                      const int* esrc, const int* edst, int E, int Nn,
                      float* w_e, float* z, unsigned* menc, float* as, float* ad,
                      hipStream_t stream) {
  int mt = (Nn + 15) / 16, nt4 = Fout / 64;
  int waves = mt * nt4;
  gemm_f32_wmma<<<(waves + 7) / 8, 256, 0, stream>>>(xin, W, h, Nn, Fin, Fout);
  node_alpha<<<(Nn + 7) / 8, 256, 0, stream>>>(h, avs, avd, as, ad, Nn, Fout);

  int ET = E + Nn;
  fill_u32<<<(Nn + 255) / 256, 256, 0, stream>>>(menc, ENC_NEG_INF, Nn);
  fill_f32<<<(Nn + 255) / 256, 256, 0, stream>>>(z, 0.f, (long long)Nn);
  long long accn = (long long)Nn * Fout;
  fill_f32<<<(unsigned)((accn + 255) / 256), 256, 0, stream>>>(accb, 0.f, accn);

  edge_max<<<(ET + 255) / 256, 256, 0, stream>>>(as, ad, esrc, edst, menc, E, Nn);
  edge_expsum<<<(ET + 255) / 256, 256, 0, stream>>>(as, ad, esrc, edst, menc, w_e, z, E, Nn);

  long long aggt = (long long)ET * (Fout >> 2);
  aggregate<<<(unsigned)((aggt + 255) / 256), 256, 0, stream>>>(h, esrc, edst, w_e, z, accb,
                                                                E, Nn, Fout);
  bias_act<<<(unsigned)((accn + 255) / 256), 256, 0, stream>>>(accb, bias, xout, accn, Fout,
                                                               do_relu);
}

extern "C" void kernel_launch(void* const* d_in, const int* in_sizes, int n_in,
                              void* d_out, int out_size, void* d_ws, size_t ws_size,
                              hipStream_t stream) {
  const float* x   = (const float*)d_in[0];
  const int*  eidx = (const int*)d_in[1];
  const float* W1  = (const float*)d_in[2];
  const float* a1s = (const float*)d_in[3];
  const float* a1d = (const float*)d_in[4];
  const float* b1  = (const float*)d_in[5];
  const float* W2  = (const float*)d_in[6];
  const float* a2s = (const float*)d_in[7];
  const float* a2d = (const float*)d_in[8];
  const float* b2  = (const float*)d_in[9];
  float* out = (float*)d_out;

  const int IN_C = 128, HID_C = 256, OUT_C = 128;
  int Nn = in_sizes[0] / IN_C;
  int E  = in_sizes[1] / 2;
  const int* esrc = eidx;
  const int* edst = eidx + E;
  int ET = E + Nn;

  // workspace layout (floats)
  float* bufA = (float*)d_ws;                       // Nn*256 : h1, then h2 + acc2
  float* bufB = bufA + (size_t)Nn * HID_C;          // Nn*256 : acc1 -> relu(x2) in place
  float* w_e  = bufB + (size_t)Nn * HID_C;          // ET
  float* z    = w_e + ET;                           // Nn
  unsigned* menc = (unsigned*)(z + Nn);             // Nn
  float* as   = (float*)(menc + Nn);                // Nn
  float* ad   = as + Nn;                            // Nn
  (void)ws_size; (void)n_in; (void)out_size;

  // Layer 1: x[Nn,128] -> h1[Nn,256]; aggregate -> bufB; relu in place
  run_layer(x, IN_C, W1, a1s, a1d, b1,
            /*h=*/bufA, /*acc=*/bufB, /*xout=*/bufB, HID_C, /*relu=*/1,
            esrc, edst, E, Nn, w_e, z, menc, as, ad, stream);

  // Layer 2: bufB[Nn,256] -> h2[Nn,128] (reuse bufA front), acc2 in bufA back, out -> d_out
  float* h2   = bufA;
  float* acc2 = bufA + (size_t)Nn * OUT_C;
  run_layer(bufB, HID_C, W2, a2s, a2d, b2,
            /*h=*/h2, /*acc=*/acc2, /*xout=*/out, OUT_C, /*relu=*/0,
            esrc, edst, E, Nn, w_e, z, menc, as, ad, stream);
}